// CoAttention_33526514713187
// MI455X (gfx1250) — compile-verified
//
#include <hip/hip_runtime.h>

typedef float v2f __attribute__((ext_vector_type(2)));
typedef float v8f __attribute__((ext_vector_type(8)));

#define B_   16
#define P_   196   // H*W = 14*14
#define D_   512
#define L_   32
#define HID_ 512
#define A_   512

// ---------------------------------------------------------------------------
// Kernel 1: hlin[b,l,a] = sum_k hiddens[b,l,k] * W_hidden[a,k] + b_hidden[a]
// One wave (32 threads) computes one 16x16 tile via V_WMMA_F32_16X16X4_F32.
// ---------------------------------------------------------------------------
__global__ __launch_bounds__(32) void hlin_kernel(const float* __restrict__ hiddens,
                                                  const float* __restrict__ W_hidden,
                                                  const float* __restrict__ b_hidden,
                                                  float* __restrict__ hlin) {
    int blk = blockIdx.x;
    int b   = blk >> 6;          // 64 tiles per batch: 2 (L) x 32 (A)
    int rem = blk & 63;
    int m0  = (rem >> 5) * 16;   // row tile in L
    int n0  = (rem & 31) * 16;   // col tile in A
    int lane = threadIdx.x;
    int hl   = lane >> 4;        // 0/1: K half selector
    int mr   = lane & 15;        // M row (A-operand) / N col (B,C,D)

    const float* Abase = hiddens  + ((size_t)b * L_ + m0 + mr) * HID_;
    const float* Bbase = W_hidden + (size_t)(n0 + mr) * HID_;   // row = output col a

    float bias = b_hidden[n0 + mr];
    v8f c;
#pragma unroll
    for (int r = 0; r < 8; ++r) c[r] = bias;

    for (int k = 0; k < HID_; k += 4) {
        int kk = k + 2 * hl;                       // VGPR0: K=kk, VGPR1: K=kk+1
        float2 av = *(const float2*)(Abase + kk);
        float2 bv = *(const float2*)(Bbase + kk);
        v2f a;  a[0] = av.x;  a[1] = av.y;
        v2f bb; bb[0] = bv.x; bb[1] = bv.y;
        c = __builtin_amdgcn_wmma_f32_16x16x4_f32(false, a, false, bb,
                                                  (short)0, c, false, false);
    }

    float* O = hlin + (size_t)b * L_ * A_;
#pragma unroll
    for (int r = 0; r < 8; ++r) {
        int m = m0 + r + 8 * hl;
        O[(size_t)m * A_ + n0 + mr] = c[r];
    }
}

// ---------------------------------------------------------------------------
// Kernel 2: scores[b,l,p] = sum_a relu(fmaps[b,p,a] + hlin[b,l,a])*W_rect[a] + b_rect
// Block caches 16 hlin rows (32KB LDS) + W_rect; each pixel row of fmaps is
// loaded once into registers and reused across all 16 l values.
// ---------------------------------------------------------------------------
__global__ __launch_bounds__(256) void score_kernel(const float* __restrict__ maps,
                                                    const float* __restrict__ hlin,
                                                    const float* __restrict__ W_rect,
                                                    const float* __restrict__ b_rect,
                                                    float* __restrict__ scores) {
    __shared__ float s_h[16][A_];
    __shared__ float s_w[A_];

    int b   = blockIdx.x;      // 0..15
    int lc  = blockIdx.y;      // 0..1 : which 16 l's
    int pc  = blockIdx.z;      // 0..3 : which 49 pixels
    int tid = threadIdx.x;
    int l0  = lc * 16;

    const float* hb = hlin + ((size_t)b * L_ + l0) * A_;   // 16 contiguous rows
    for (int idx = tid; idx < 16 * A_; idx += 256)
        s_h[idx >> 9][idx & (A_ - 1)] = hb[idx];
    for (int i = tid; i < A_; i += 256)
        s_w[i] = W_rect[i];
    __syncthreads();

    int wave = tid >> 5, lane = tid & 31;
    float br = b_rect[0];
    int p0 = pc * 49;

    for (int p = p0 + wave; p < p0 + 49; p += 8) {
        const float* frow = maps + ((size_t)b * P_ + p) * D_;
        float f[16];
#pragma unroll
        for (int j = 0; j < 16; ++j) f[j] = frow[lane + 32 * j];

        for (int lp = 0; lp < 16; ++lp) {
            float s = 0.f;
#pragma unroll
            for (int j = 0; j < 16; ++j) {
                int a = lane + 32 * j;
                float v = f[j] + s_h[lp][a];
                s += fmaxf(v, 0.f) * s_w[a];
            }
#pragma unroll
            for (int off = 16; off > 0; off >>= 1)
                s += __shfl_xor(s, off, 32);
            if (lane == 0)
                scores[((size_t)b * L_ + l0 + lp) * P_ + p] = s + br;
        }
    }
}

// ---------------------------------------------------------------------------
// Kernel 3: softmax over P=196 per (b,l); writes attn into d_out slice.
// ---------------------------------------------------------------------------
__global__ __launch_bounds__(256) void softmax_kernel(const float* __restrict__ scores,
                                                      float* __restrict__ attn) {
    __shared__ float s_red[256];
    int row = blockIdx.x;            // b*L + l
    int tid = threadIdx.x;
    const float* srow = scores + (size_t)row * P_;

    float v = (tid < P_) ? srow[tid] : -3.402823e38f;
    s_red[tid] = v;
    __syncthreads();
    for (int off = 128; off > 0; off >>= 1) {
        if (tid < off) s_red[tid] = fmaxf(s_red[tid], s_red[tid + off]);
        __syncthreads();
    }
    float mx = s_red[0];
    __syncthreads();

    float e = (tid < P_) ? __expf(v - mx) : 0.f;
    s_red[tid] = e;
    __syncthreads();
    for (int off = 128; off > 0; off >>= 1) {
        if (tid < off) s_red[tid] += s_red[tid + off];
        __syncthreads();
    }
    float inv = 1.f / s_red[0];
    if (tid < P_) attn[(size_t)row * P_ + tid] = e * inv;
}

// ---------------------------------------------------------------------------
// Kernel 4: ctx[b,l,d] = sum_p attn[b,l,p] * fmaps[b,p,d]   (K = 196 = 49*4)
// ---------------------------------------------------------------------------
__global__ __launch_bounds__(32) void ctx_kernel(const float* __restrict__ attn,
                                                 const float* __restrict__ maps,
                                                 float* __restrict__ ctx) {
    int blk = blockIdx.x;
    int b   = blk >> 6;
    int rem = blk & 63;
    int m0  = (rem >> 5) * 16;   // L tile
    int n0  = (rem & 31) * 16;   // D tile
    int lane = threadIdx.x;
    int hl   = lane >> 4;
    int mr   = lane & 15;

    const float* Abase = attn + ((size_t)b * L_ + m0 + mr) * P_;
    const float* Bbase = maps + (size_t)b * P_ * D_ + n0 + mr;  // column n0+mr

    v8f c;
#pragma unroll
    for (int r = 0; r < 8; ++r) c[r] = 0.f;

    for (int k = 0; k < P_; k += 4) {
        int kk = k + 2 * hl;
        float2 av = *(const float2*)(Abase + kk);
        v2f a;  a[0] = av.x; a[1] = av.y;
        v2f bb;
        bb[0] = Bbase[(size_t)kk * D_];
        bb[1] = Bbase[(size_t)(kk + 1) * D_];
        c = __builtin_amdgcn_wmma_f32_16x16x4_f32(false, a, false, bb,
                                                  (short)0, c, false, false);
    }

    float* O = ctx + (size_t)b * L_ * D_;
#pragma unroll
    for (int r = 0; r < 8; ++r)
        O[(size_t)(m0 + r + 8 * hl) * D_ + n0 + mr] = c[r];
}

// ---------------------------------------------------------------------------
// Kernel 5: co_att[b,l,h] = (sum_d ctx[b,l,d]*W_co[h,d] + b_co[h]) * hiddens[b,l,h]
// ---------------------------------------------------------------------------
__global__ __launch_bounds__(32) void coatt_kernel(const float* __restrict__ ctx,
                                                   const float* __restrict__ W_co,
                                                   const float* __restrict__ b_co,
                                                   const float* __restrict__ hiddens,
                                                   float* __restrict__ out) {
    int blk = blockIdx.x;
    int b   = blk >> 6;
    int rem = blk & 63;
    int m0  = (rem >> 5) * 16;   // L tile
    int n0  = (rem & 31) * 16;   // HID tile
    int lane = threadIdx.x;
    int hl   = lane >> 4;
    int mr   = lane & 15;

    const float* Abase = ctx  + ((size_t)b * L_ + m0 + mr) * D_;
    const float* Bbase = W_co + (size_t)(n0 + mr) * D_;   // row = output col h

    float bias = b_co[n0 + mr];
    v8f c;
#pragma unroll
    for (int r = 0; r < 8; ++r) c[r] = bias;

    for (int k = 0; k < D_; k += 4) {
        int kk = k + 2 * hl;
        float2 av = *(const float2*)(Abase + kk);
        float2 bv = *(const float2*)(Bbase + kk);
        v2f a;  a[0] = av.x;  a[1] = av.y;
        v2f bb; bb[0] = bv.x; bb[1] = bv.y;
        c = __builtin_amdgcn_wmma_f32_16x16x4_f32(false, a, false, bb,
                                                  (short)0, c, false, false);
    }

    const float* hbase = hiddens + (size_t)b * L_ * HID_;
    float* O = out + (size_t)b * L_ * HID_;
#pragma unroll
    for (int r = 0; r < 8; ++r) {
        int m = m0 + r + 8 * hl;
        O[(size_t)m * HID_ + n0 + mr] = c[r] * hbase[(size_t)m * HID_ + n0 + mr];
    }
}

// ---------------------------------------------------------------------------
extern "C" void kernel_launch(void* const* d_in, const int* in_sizes, int n_in,
                              void* d_out, int out_size, void* d_ws, size_t ws_size,
                              hipStream_t stream) {
    const float* maps     = (const float*)d_in[0];
    const float* hiddens  = (const float*)d_in[1];
    const float* W_hidden = (const float*)d_in[2];
    const float* b_hidden = (const float*)d_in[3];
    const float* W_rect   = (const float*)d_in[4];
    const float* b_rect   = (const float*)d_in[5];
    const float* W_co     = (const float*)d_in[6];
    const float* b_co     = (const float*)d_in[7];

    float* out  = (float*)d_out;
    float* co   = out;                                   // B*L*HID
    float* attn = out + (size_t)B_ * L_ * HID_;          // B*L*P

    float* hlin   = (float*)d_ws;                        // B*L*A   (1 MB)
    float* ctx    = hlin + (size_t)B_ * L_ * A_;         // B*L*D   (1 MB)
    float* scores = ctx  + (size_t)B_ * L_ * D_;         // B*L*P   (0.4 MB)

    // 1) hlin = hiddens @ W_hidden^T + b_hidden        (WMMA f32)
    hlin_kernel<<<B_ * 2 * 32, 32, 0, stream>>>(hiddens, W_hidden, b_hidden, hlin);

    // 2) scores = ReLU(fmaps + hlin) . W_rect + b_rect (LDS-tiled)
    score_kernel<<<dim3(B_, 2, 4), 256, 0, stream>>>(maps, hlin, W_rect, b_rect, scores);

    // 3) attn = softmax_P(scores)  -> d_out slice
    softmax_kernel<<<B_ * L_, 256, 0, stream>>>(scores, attn);

    // 4) ctx = attn @ fmaps                            (WMMA f32, K=196)
    ctx_kernel<<<B_ * 2 * 32, 32, 0, stream>>>(attn, maps, ctx);

    // 5) co_att = (ctx @ W_co^T + b_co) * hiddens      (WMMA f32)
    coatt_kernel<<<B_ * 2 * 32, 32, 0, stream>>>(ctx, W_co, b_co, hiddens, co);
}